// MelSpecFitness_32933809226295
// MI455X (gfx1250) — compile-verified
//
#include <hip/hip_runtime.h>
#include <hip/hip_bf16.h>
#include <math.h>

// ---------------- problem constants ----------------
#define SR        48000
#define N_FFT     2048
#define LOG2_NFFT 11
#define HOP       64
#define N_MELS    128
#define N_FREQ    1025            // N_FFT/2 + 1
#define KPAD      1028            // N_FREQ padded to multiple of 4 (WMMA K step)
#define N_TGT     64
#define BATCH     32
#define NFRAMES   751             // 1 + 48000/64
#define TOTFRAMES (BATCH * NFRAMES)   // 24032 = 1502 * 16
#define FR_PER_BLK 16
#define TCHUNK    32
#define NCHUNK    ((NFRAMES + TCHUNK - 1) / TCHUNK)  // 24

typedef __attribute__((ext_vector_type(2))) float v2f;
typedef __attribute__((ext_vector_type(8))) float v8f;

// =====================================================================
// Kernel 0 (one-shot, tiny): fbT[m][k] = fb[k][m], K zero-padded to 1028.
// Gives the WMMA K-loop contiguous per-lane B fragments (global_load_b64)
// and removes every bounds check from the hot loop.
// =====================================================================
__global__ __launch_bounds__(256)
void transpose_fb_kernel(const float* __restrict__ fb,   // [1025][128]
                         float* __restrict__ fbT)        // [128][1028]
{
    int idx = blockIdx.x * 256 + threadIdx.x;
    if (idx >= N_MELS * KPAD) return;
    int m = idx / KPAD;
    int k = idx - m * KPAD;
    fbT[idx] = (k < N_FREQ) ? fb[(size_t)k * N_MELS + m] : 0.0f;
}

// =====================================================================
// Kernel 1: 16 frames/block. Window + 2048-pt radix-2 FFT in LDS, power
// spectrum into LDS, then mel projection via V_WMMA_F32_16X16X4_F32.
// mel_ws layout: [g][m] where g = b*751 + t   (g-major, 128 mels contiguous)
// =====================================================================
__global__ __launch_bounds__(256)
void melspec_fft_wmma_kernel(const float* __restrict__ x,
                             const float* __restrict__ fbT,  // [128][1028]
                             float* __restrict__ mel_ws)
{
    __shared__ float re[N_FFT];
    __shared__ float im[N_FFT];
    __shared__ float powerLDS[FR_PER_BLK * KPAD];   // 16 x 1028, zero-padded K

    const int tid  = threadIdx.x;
    const int g0   = blockIdx.x * FR_PER_BLK;

    // ---------------- FFT phase: 16 frames sequentially ----------------
    for (int fi = 0; fi < FR_PER_BLK; ++fi) {
        const int g = g0 + fi;
        const int b = g / NFRAMES;
        const int t = g - b * NFRAMES;
        const float* xb = x + (size_t)b * SR;

        // load frame (reflect pad), Hann window, bit-reversed into LDS
        for (int n = tid; n < N_FFT; n += 256) {
            int j = t * HOP + n - (N_FFT / 2);          // position in signal
            j = (j < 0) ? -j : j;
            j = (j >= SR) ? (2 * SR - 2 - j) : j;
            float w = 0.5f * (1.0f - __cosf(6.2831853071795864f *
                                            (float)n / (float)N_FFT));
            int r = (int)(__brev((unsigned)n) >> (32 - LOG2_NFFT));
            re[r] = xb[j] * w;
            im[r] = 0.0f;
        }
        __syncthreads();

        // radix-2 DIT, 11 stages, 1024 butterflies/stage, 4 per thread
        for (int s = 1; s <= LOG2_NFFT; ++s) {
            const int m    = 1 << s;
            const int half = m >> 1;
            for (int k = tid; k < (N_FFT / 2); k += 256) {
                int pos = k & (half - 1);
                int grp = k >> (s - 1);
                int i0  = (grp << s) + pos;
                int i1  = i0 + half;
                float ang = -6.2831853071795864f * (float)pos / (float)m;
                float wi_s, wr_c;
                __sincosf(ang, &wi_s, &wr_c);
                float r1 = re[i1], q1 = im[i1];
                float tr = wr_c * r1 - wi_s * q1;
                float ti = wr_c * q1 + wi_s * r1;
                float r0 = re[i0], q0 = im[i0];
                re[i1] = r0 - tr;  im[i1] = q0 - ti;
                re[i0] = r0 + tr;  im[i0] = q0 + ti;
            }
            __syncthreads();
        }

        // power spectrum -> LDS row fi (pad K 1025..1027 with zeros)
        for (int f = tid; f < KPAD; f += 256) {
            float pv = 0.0f;
            if (f <= N_FFT / 2) pv = re[f] * re[f] + im[f] * im[f];
            powerLDS[fi * KPAD + f] = pv;
        }
        __syncthreads();
    }

    // ---------------- WMMA phase: D(16x128) = power(16x1028) * fbT^T ---
    const int lane = tid & 31;
    const int wave = tid >> 5;          // 8 waves -> 8 mel tiles of 16
    const int n0   = wave * 16;
    const int Ma   = lane & 15;         // A row (frame)
    const int kofs = (lane >> 4) << 1;  // lanes 16-31 carry K+2,K+3
    const int nc   = lane & 15;         // B/D column within tile

    // per-lane base pointers: both 8-byte aligned, fully unconditional
    const float* prow  = powerLDS + Ma * KPAD + kofs;             // LDS
    const float* fbcol = fbT + (size_t)(n0 + nc) * KPAD + kofs;   // global

    v8f acc0 = {};
    v8f acc1 = {};
    #pragma unroll 2
    for (int f = 0; f < 1024; f += 8) {
        v2f a0 = *(const v2f*)(prow  + f);
        v2f b0 = *(const v2f*)(fbcol + f);
        v2f a1 = *(const v2f*)(prow  + f + 4);
        v2f b1 = *(const v2f*)(fbcol + f + 4);
        acc0 = __builtin_amdgcn_wmma_f32_16x16x4_f32(
                   false, a0, false, b0, (short)0, acc0, false, false);
        acc1 = __builtin_amdgcn_wmma_f32_16x16x4_f32(
                   false, a1, false, b1, (short)0, acc1, false, false);
    }
    {   // tail: K = 1024..1027 (zero-padded on both operands)
        v2f a = *(const v2f*)(prow  + 1024);
        v2f b = *(const v2f*)(fbcol + 1024);
        acc0 = __builtin_amdgcn_wmma_f32_16x16x4_f32(
                   false, a, false, b, (short)0, acc0, false, false);
    }
    v8f acc = acc0 + acc1;

    // D layout: reg r, lanes 0-15 -> M=r, lanes 16-31 -> M=r+8
    const int mrow = (lane >> 4) << 3;
    #pragma unroll
    for (int r = 0; r < 8; ++r) {
        int M = r + mrow;
        mel_ws[(size_t)(g0 + M) * N_MELS + n0 + nc] = acc[r];
    }
}

// =====================================================================
// Kernel 2: per (b, t-chunk) block: load mel chunk to LDS (transposed),
// accumulate sum|mel - tgt[j]| for all 64 targets, write partial sums.
// partials layout: [b][chunk][j]
// =====================================================================
__global__ __launch_bounds__(256)
void absdiff_partial_kernel(const float* __restrict__ mel_ws,   // [g][m]
                            const float* __restrict__ tgt,      // [j][m][t]
                            float* __restrict__ partials)
{
    __shared__ float melC[N_MELS * TCHUNK];   // [m][tt] transposed, 16 KB
    __shared__ float red[8];

    const int b     = blockIdx.x;
    const int chunk = blockIdx.y;
    const int t0    = chunk * TCHUNK;
    const int tid   = threadIdx.x;
    const int lane  = tid & 31;
    const int wave  = tid >> 5;

    // load chunk: coalesced over m from mel_ws, transpose into LDS
    for (int idx = tid; idx < N_MELS * TCHUNK; idx += 256) {
        int tt = idx >> 7;          // 0..31
        int m  = idx & 127;
        int t  = t0 + tt;
        float v = 0.0f;
        if (t < NFRAMES)
            v = mel_ws[(size_t)(b * NFRAMES + t) * N_MELS + m];
        melC[m * TCHUNK + tt] = v;
    }
    __syncthreads();

    const int  t  = t0 + lane;
    const bool tv = (t < NFRAMES);

    for (int j = 0; j < N_TGT; ++j) {
        const float* tj = tgt + (size_t)j * N_MELS * NFRAMES;
        // warm L2/L0 for the next target while we crunch this one
        if (j + 1 < N_TGT && tv)
            __builtin_prefetch(tgt + (size_t)(j + 1) * N_MELS * NFRAMES +
                               (size_t)(wave * 16) * NFRAMES + t, 0, 0);
        float s = 0.0f;
        #pragma unroll 4
        for (int mi = 0; mi < 16; ++mi) {
            int m = wave * 16 + mi;                 // wave owns 16 mel rows
            if (tv) {
                float tv2 = tj[(size_t)m * NFRAMES + t];  // coalesced over t
                float mv  = melC[m * TCHUNK + lane];
                s += fabsf(mv - tv2);
            }
        }
        // reduce 256 -> 1
        #pragma unroll
        for (int off = 16; off; off >>= 1)
            s += __shfl_down(s, off, 32);
        if (lane == 0) red[wave] = s;
        __syncthreads();
        if (tid == 0) {
            float tot = 0.0f;
            #pragma unroll
            for (int w = 0; w < 8; ++w) tot += red[w];
            partials[((size_t)b * NCHUNK + chunk) * N_TGT + j] = tot;
        }
        __syncthreads();
    }
}

// =====================================================================
// Kernel 3: per b: sum chunks per target, scale to mean, min over targets
// =====================================================================
__global__ __launch_bounds__(64)
void min_reduce_kernel(const float* __restrict__ partials,
                       float* __restrict__ out)
{
    __shared__ float r2[2];
    const int b   = blockIdx.x;
    const int j   = threadIdx.x;       // 64 threads = 64 targets
    const int lane = j & 31;
    const int wid  = j >> 5;

    float s = 0.0f;
    for (int c = 0; c < NCHUNK; ++c)
        s += partials[((size_t)b * NCHUNK + c) * N_TGT + j];
    s *= (1.0f / ((float)N_MELS * (float)NFRAMES));

    #pragma unroll
    for (int off = 16; off; off >>= 1)
        s = fminf(s, __shfl_down(s, off, 32));
    if (lane == 0) r2[wid] = s;
    __syncthreads();
    if (j == 0) out[b] = fminf(r2[0], r2[1]);
}

// =====================================================================
extern "C" void kernel_launch(void* const* d_in, const int* in_sizes, int n_in,
                              void* d_out, int out_size, void* d_ws, size_t ws_size,
                              hipStream_t stream)
{
    const float* x   = (const float*)d_in[0];   // [32][48000]
    const float* tgt = (const float*)d_in[1];   // [64][128][751]
    const float* fb  = (const float*)d_in[2];   // [1025][128]
    float* out = (float*)d_out;                 // [32]

    float* mel_ws   = (float*)d_ws;                                // 24032*128
    float* partials = mel_ws + (size_t)TOTFRAMES * N_MELS;         // 32*24*64
    float* fbT      = partials + (size_t)BATCH * NCHUNK * N_TGT;   // 128*1028

    transpose_fb_kernel<<<(N_MELS * KPAD + 255) / 256, 256, 0, stream>>>(fb, fbT);

    melspec_fft_wmma_kernel<<<TOTFRAMES / FR_PER_BLK, 256, 0, stream>>>(
        x, fbT, mel_ws);

    dim3 grid2(BATCH, NCHUNK);
    absdiff_partial_kernel<<<grid2, 256, 0, stream>>>(mel_ws, tgt, partials);

    min_reduce_kernel<<<BATCH, 64, 0, stream>>>(partials, out);
}